// MEHGNN_lp1_32804960207438
// MI455X (gfx1250) — compile-verified
//
#include <hip/hip_runtime.h>
#include <math.h>

#define HID   64
#define D2    32
#define NHEADS 8
#define NG    50000
#define EKG   400000
#define EMP   200000
#define BATCH 4096
#define ENTN  60000
#define NCVE  50000
#define NPVS  30000
#define AVD   128
#define OUTD  64
#define FEATD (NHEADS * HID)   // 512

typedef __attribute__((ext_vector_type(2))) float v2f;
typedef __attribute__((ext_vector_type(8))) float v8f;

// ---- order-preserving float <-> uint encoding for atomic segment-max ----
__device__ __forceinline__ unsigned f2o(float f) {
  unsigned u = __float_as_uint(f);
  return (u & 0x80000000u) ? ~u : (u | 0x80000000u);
}
__device__ __forceinline__ float o2f(unsigned u) {
  return __uint_as_float((u & 0x80000000u) ? (u ^ 0x80000000u) : ~u);
}
#define ORD_NEG_INF 0x007FFFFFu   // f2o(-inf)

// ---------------- RGAT ----------------

// Wr[et][i] = sum_j W[i][j] * rel[et][j]   (W: [128,64] row-major, rel: [20,64])
__global__ void k_wr(const float* __restrict__ W, const float* __restrict__ rel,
                     float* __restrict__ Wr) {
  int t = blockIdx.x * blockDim.x + threadIdx.x;
  if (t >= 20 * 128) return;
  int et = t >> 7, i = t & 127;
  const float* w = W + i * HID;
  const float* r = rel + et * HID;
  float s = 0.f;
#pragma unroll
  for (int j = 0; j < HID; j++) s += w[j] * r[j];
  Wr[et * 128 + i] = s;
}

__global__ void k_init_sm(unsigned* __restrict__ nmax, float* __restrict__ nsum, int n) {
  int t = blockIdx.x * blockDim.x + threadIdx.x;
  if (t < n) { nmax[t] = ORD_NEG_INF; nsum[t] = 0.f; }
}

__global__ void k_rgat_edge(const float* __restrict__ ent, const float* __restrict__ Wr,
                            const int* __restrict__ head, const int* __restrict__ tail,
                            const int* __restrict__ etype, float* __restrict__ ebuf,
                            unsigned* __restrict__ nmax) {
  int e = blockIdx.x * blockDim.x + threadIdx.x;
  if (e >= EKG) return;
  int h = head[e], t = tail[e], et = etype[e];
  const float4* eh = (const float4*)(ent + (size_t)h * HID);
  const float4* tl = (const float4*)(ent + (size_t)t * HID);
  const float4* w  = (const float4*)(Wr + et * 128);
  float s = 0.f;
#pragma unroll
  for (int k = 0; k < 16; k++) {
    float4 a = eh[k], b = w[k];
    s += a.x * b.x + a.y * b.y + a.z * b.z + a.w * b.w;
  }
#pragma unroll
  for (int k = 0; k < 16; k++) {
    float4 a = tl[k], b = w[16 + k];
    s += a.x * b.x + a.y * b.y + a.z * b.z + a.w * b.w;
  }
  float ev = s > 0.f ? s : 0.2f * s;          // leaky_relu 0.2
  ebuf[e] = ev;
  atomicMax(&nmax[h], f2o(ev));
}

__global__ void k_rgat_exp(float* __restrict__ ebuf, const int* __restrict__ head,
                           const unsigned* __restrict__ nmax, float* __restrict__ nsum) {
  int e = blockIdx.x * blockDim.x + threadIdx.x;
  if (e >= EKG) return;
  int h = head[e];
  float ex = expf(ebuf[e] - o2f(nmax[h]));
  ebuf[e] = ex;
  atomicAdd(&nsum[h], ex);
}

__global__ void k_rgat_scatter(const float* __restrict__ ent, const float* __restrict__ ebuf,
                               const float* __restrict__ nsum,
                               const int* __restrict__ head, const int* __restrict__ tail,
                               float* __restrict__ out) {
  long long gid = (long long)blockIdx.x * blockDim.x + threadIdx.x;
  if (gid >= (long long)EKG * 16) return;
  int e = (int)(gid >> 4);
  int f = ((int)gid & 15) * 4;
  int h = head[e], t = tail[e];
  float alpha = ebuf[e] / nsum[h];
  const float* src = ent + (size_t)t * HID + f;
  float* dst = out + (size_t)h * HID + f;
  atomicAdd(dst + 0, alpha * src[0]);
  atomicAdd(dst + 1, alpha * src[1]);
  atomicAdd(dst + 2, alpha * src[2]);
  atomicAdd(dst + 3, alpha * src[3]);
}

// ---------------- RotatE relation vectors ----------------

__global__ void k_rfull(const float* __restrict__ r_vec, float* __restrict__ rfull) {
  int t = blockIdx.x * blockDim.x + threadIdx.x;
  if (t >= 2 * D2) return;
  int p = t / D2, k = t % D2;
  float re = r_vec[(p * D2 + k) * 2 + 0], im = r_vec[(p * D2 + k) * 2 + 1];
  float inv = rsqrtf(re * re + im * im + 1e-12f);
  re *= inv; im *= inv;
  float* r0 = rfull + ((2 * p) * D2 + k) * 2;
  float* r1 = rfull + ((2 * p + 1) * D2 + k) * 2;
  r0[0] = re; r0[1] = im;
  r1[0] = re; r1[1] = -im;   // conjugate
}

// Fmp[mp][l][k] cumulative rotations: F[0]=r(et0)*r(et1), F[1]=r(et1), F[2]=1
__global__ void k_fmp(const float* __restrict__ rfull, float* __restrict__ Fmp) {
  int t = blockIdx.x * blockDim.x + threadIdx.x;
  if (t >= 4 * D2) return;
  int mp = t / D2, k = t % D2;
  const int ET[4][2] = {{0, 1}, {2, 3}, {1, 0}, {3, 2}};
  int e0 = ET[mp][0], e1 = ET[mp][1];
  float r1re = rfull[(e1 * D2 + k) * 2], r1im = rfull[(e1 * D2 + k) * 2 + 1];
  float r0re = rfull[(e0 * D2 + k) * 2], r0im = rfull[(e0 * D2 + k) * 2 + 1];
  float f0re = r1re * r0re - r1im * r0im;
  float f0im = r1re * r0im + r1im * r0re;
  float* F = Fmp + ((mp * 3 + 0) * D2 + k) * 2; F[0] = f0re; F[1] = f0im;
  F = Fmp + ((mp * 3 + 1) * D2 + k) * 2;        F[0] = r1re; F[1] = r1im;
  F = Fmp + ((mp * 3 + 2) * D2 + k) * 2;        F[0] = 1.f;  F[1] = 0.f;
}

__global__ void k_needed(const int* __restrict__ tgt, int* __restrict__ needed) {
  int b = blockIdx.x * blockDim.x + threadIdx.x;
  if (b < BATCH) needed[tgt[b]] = 1;
}

// ---------------- Metapath ----------------

__global__ void k_mp_edge(const float* __restrict__ ent, const float* __restrict__ pvs,
                          const int* __restrict__ idx, const int* __restrict__ dst,
                          const float* __restrict__ Fmp3, const float* __restrict__ attn,
                          const int* __restrict__ needed,
                          float* __restrict__ hidden, float* __restrict__ abuf,
                          unsigned* __restrict__ am) {
  int e = blockIdx.x * blockDim.x + threadIdx.x;
  if (e >= EMP) return;
  int d = dst[e];
  if (!needed[d]) return;   // dst never reaches tgt: contributes nothing downstream
  float hid[HID];
#pragma unroll
  for (int k = 0; k < HID; k++) hid[k] = 0.f;
#pragma unroll
  for (int l = 0; l < 3; l++) {
    int node = idx[(size_t)e * 3 + l];
    const float* v = (node < NCVE) ? (ent + (size_t)node * HID)
                                   : (pvs + (size_t)(node - NCVE) * HID);
    const float* F = Fmp3 + l * D2 * 2;
#pragma unroll
    for (int k = 0; k < D2; k++) {
      float vr = v[2 * k], vi = v[2 * k + 1];
      float fr = F[2 * k], fi = F[2 * k + 1];
      hid[2 * k]     += vr * fr - vi * fi;
      hid[2 * k + 1] += vr * fi + vi * fr;
    }
  }
  const float inv3 = 1.f / 3.f;
  float* ho = hidden + (size_t)e * HID;
#pragma unroll
  for (int k = 0; k < HID; k++) { hid[k] *= inv3; ho[k] = hid[k]; }
#pragma unroll
  for (int h = 0; h < NHEADS; h++) {
    const float* a = attn + h * HID;
    float s = 0.f;
#pragma unroll
    for (int k = 0; k < HID; k++) s += hid[k] * a[k];
    float av = s > 0.f ? s : 0.01f * s;       // leaky_relu 0.01
    abuf[(size_t)e * NHEADS + h] = av;
    atomicMax(&am[(size_t)d * NHEADS + h], f2o(av));
  }
}

__global__ void k_mp_exp(float* __restrict__ abuf, const int* __restrict__ dst,
                         const int* __restrict__ needed,
                         const unsigned* __restrict__ am, float* __restrict__ asum) {
  long long g = (long long)blockIdx.x * blockDim.x + threadIdx.x;
  if (g >= (long long)EMP * NHEADS) return;
  int e = (int)(g / NHEADS), h = (int)(g % NHEADS);
  int d = dst[e];
  if (!needed[d]) return;
  float ex = expf(abuf[g] - o2f(am[(size_t)d * NHEADS + h]));
  abuf[g] = ex;
  atomicAdd(&asum[(size_t)d * NHEADS + h], ex);
}

__global__ void k_mp_scatter(const float* __restrict__ hidden, const float* __restrict__ abuf,
                             const float* __restrict__ asum, const int* __restrict__ dst,
                             const int* __restrict__ needed, float* __restrict__ ret) {
  long long g = (long long)blockIdx.x * blockDim.x + threadIdx.x;
  if (g >= (long long)EMP * NHEADS) return;
  int e = (int)(g / NHEADS), h = (int)(g % NHEADS);
  int d = dst[e];
  if (!needed[d]) return;
  float alpha = abuf[g] / asum[(size_t)d * NHEADS + h];
  const float* src = hidden + (size_t)e * HID;
  float* r = ret + ((size_t)d * NHEADS + h) * HID;
#pragma unroll
  for (int k = 0; k < HID; k++) atomicAdd(&r[k], alpha * src[k]);
}

__global__ void k_mp_gather(const float* __restrict__ ret, const int* __restrict__ tgt,
                            float* __restrict__ hs) {
  int g = blockIdx.x * blockDim.x + threadIdx.x;
  if (g >= BATCH * FEATD) return;
  int b = g / FEATD, j = g % FEATD;
  float x = ret[(size_t)tgt[b] * FEATD + j];
  hs[g] = x > 0.f ? x : (expf(x) - 1.f);      // elu
}

// ---------------- WMMA fp32 GEMM: C[M,N] = A[M,K] @ Wt[N,K]^T (+bias) ----------------
// One wave owns one 16x16 C tile; V_WMMA_F32_16X16X4_F32 over K in steps of 4.
// If C!=null: write C (+bias). Else: epilogue tanh(.+bias) summed per column into colsum.
__global__ void k_gemm_wmma(const float* __restrict__ A, const float* __restrict__ Wt,
                            const float* __restrict__ bias, float* __restrict__ C,
                            float* __restrict__ colsum, int M, int N, int K) {
  int wave = (int)((blockIdx.x * blockDim.x + threadIdx.x) >> 5);
  int lane = threadIdx.x & 31;
  int ntiles = N >> 4;
  int mtiles = M >> 4;
  if (wave >= mtiles * ntiles) return;        // uniform per wave: EXEC stays all-1s
  int tm = wave / ntiles, tn = wave % ntiles;
  int r16 = lane & 15;
  int row = tm * 16 + r16;                    // A: M = lane%16
  int col = tn * 16 + r16;                    // B/C: N = lane%16
  int khalf = (lane < 16) ? 0 : 2;            // K pair per lane half
  const float* arow = A + (size_t)row * K;
  const float* brow = Wt + (size_t)col * K;   // B[k][n] = Wt[n][k]
  v8f c = {0.f, 0.f, 0.f, 0.f, 0.f, 0.f, 0.f, 0.f};
  for (int k0 = 0; k0 < K; k0 += 4) {
    int kk = k0 + khalf;
    v2f a, b;
    a[0] = arow[kk]; a[1] = arow[kk + 1];
    b[0] = brow[kk]; b[1] = brow[kk + 1];
    c = __builtin_amdgcn_wmma_f32_16x16x4_f32(false, a, false, b, (short)0, c,
                                              false, false);
  }
  float bv = bias ? bias[col] : 0.f;
  if (C) {
#pragma unroll
    for (int i = 0; i < 8; i++) {
      int m = tm * 16 + i + ((lane < 16) ? 0 : 8);   // C VGPR layout
      C[(size_t)m * N + col] = c[i] + bv;
    }
  } else {
    float s = 0.f;
#pragma unroll
    for (int i = 0; i < 8; i++) s += tanhf(c[i] + bv);
    atomicAdd(&colsum[col], s);
  }
}

// ---------------- Fuse ----------------

__global__ void k_betas(const float* __restrict__ colsum, const float* __restrict__ w2u,
                        const float* __restrict__ w2i, float* __restrict__ beta) {
  __shared__ float sh[AVD];
  int t = threadIdx.x;
  float b[4];
  for (int p = 0; p < 4; p++) {
    const float* w2 = (p < 2) ? w2u : w2i;
    sh[t] = colsum[p * AVD + t] * w2[t];
    __syncthreads();
    for (int s = 64; s > 0; s >>= 1) {
      if (t < s) sh[t] += sh[t + s];
      __syncthreads();
    }
    b[p] = sh[0] * (1.f / (float)BATCH);
    __syncthreads();
  }
  if (t == 0) {
    float mu = fmaxf(b[0], b[1]);
    float e0 = expf(b[0] - mu), e1 = expf(b[1] - mu);
    beta[0] = e0 / (e0 + e1); beta[1] = e1 / (e0 + e1);
    float mi = fmaxf(b[2], b[3]);
    float f0 = expf(b[2] - mi), f1 = expf(b[3] - mi);
    beta[2] = f0 / (f0 + f1); beta[3] = f1 / (f0 + f1);
  }
}

__global__ void k_fuse(const float* __restrict__ hs0, const float* __restrict__ hs1,
                       const float* __restrict__ beta, float* __restrict__ h) {
  int g = blockIdx.x * blockDim.x + threadIdx.x;
  if (g >= BATCH * FEATD) return;
  h[g] = beta[0] * hs0[g] + beta[1] * hs1[g];
}

// ---------------- launch ----------------

extern "C" void kernel_launch(void* const* d_in, const int* in_sizes, int n_in,
                              void* d_out, int out_size, void* d_ws, size_t ws_size,
                              hipStream_t stream) {
  const float* entity_emb   = (const float*)d_in[0];
  const float* relation_emb = (const float*)d_in[1];
  const float* pvs          = (const float*)d_in[2];
  const float* rgat_W       = (const float*)d_in[3];
  const float* r_vec        = (const float*)d_in[4];
  const float* attn_user    = (const float*)d_in[5];
  const float* attn_item    = (const float*)d_in[6];
  const float* fc1_w_user   = (const float*)d_in[7];
  const float* fc1_b_user   = (const float*)d_in[8];
  const float* fc2_w_user   = (const float*)d_in[9];
  const float* fc1_w_item   = (const float*)d_in[10];
  const float* fc1_b_item   = (const float*)d_in[11];
  const float* fc2_w_item   = (const float*)d_in[12];
  const float* fc_user_w    = (const float*)d_in[13];
  const float* fc_user_b    = (const float*)d_in[14];
  const float* fc_item_w    = (const float*)d_in[15];
  const float* fc_item_b    = (const float*)d_in[16];
  const int* kg_head  = (const int*)d_in[17];
  const int* kg_tail  = (const int*)d_in[18];
  const int* kg_etype = (const int*)d_in[19];
  const int* idx_user = (const int*)d_in[20];
  const int* dst_user = (const int*)d_in[21];
  const int* tgt_user = (const int*)d_in[22];
  const int* idx_item = (const int*)d_in[23];
  const int* dst_item = (const int*)d_in[24];
  const int* tgt_item = (const int*)d_in[25];
  (void)in_sizes; (void)n_in; (void)out_size;

  float* out = (float*)d_out;
  float* ws = (float*)d_ws;
  size_t off = 0;
  auto alloc = [&](size_t n) { float* p = ws + off; off += n; return p; };
  float* entA    = alloc((size_t)ENTN * HID);
  float* entB    = alloc((size_t)ENTN * HID);
  float* Wr      = alloc(20 * 128);
  float* ebuf    = alloc((size_t)EMP * NHEADS);      // >= EKG, reused
  unsigned* nmax = (unsigned*)alloc((size_t)NG * NHEADS);
  float* nsum    = alloc((size_t)NG * NHEADS);
  float* hidden  = alloc((size_t)EMP * HID);
  float* rfull   = alloc(4 * D2 * 2);
  float* Fmp     = alloc(4 * 3 * D2 * 2);
  float* ret     = alloc((size_t)NG * FEATD);
  float* hs      = alloc(4 * (size_t)BATCH * FEATD);
  float* colsum  = alloc(4 * AVD);
  float* beta    = alloc(4);
  int* neededU   = (int*)alloc(NG);
  int* neededI   = (int*)alloc(NG);
  if (off * sizeof(float) > ws_size) return;  // workspace too small

  const int TB = 256;

  // ---- RGAT: 2 hops ----
  hipMemcpyAsync(entA, entity_emb, (size_t)ENTN * HID * sizeof(float),
                 hipMemcpyDeviceToDevice, stream);
  k_wr<<<(20 * 128 + TB - 1) / TB, TB, 0, stream>>>(rgat_W, relation_emb, Wr);
  float* cur = entA;
  float* nxt = entB;
  for (int hop = 0; hop < 2; hop++) {
    k_init_sm<<<(ENTN + TB - 1) / TB, TB, 0, stream>>>(nmax, nsum, ENTN);
    k_rgat_edge<<<(EKG + TB - 1) / TB, TB, 0, stream>>>(cur, Wr, kg_head, kg_tail,
                                                        kg_etype, ebuf, nmax);
    k_rgat_exp<<<(EKG + TB - 1) / TB, TB, 0, stream>>>(ebuf, kg_head, nmax, nsum);
    hipMemcpyAsync(nxt, cur, (size_t)ENTN * HID * sizeof(float),
                   hipMemcpyDeviceToDevice, stream);   // residual base
    k_rgat_scatter<<<(int)(((long long)EKG * 16 + TB - 1) / TB), TB, 0, stream>>>(
        cur, ebuf, nsum, kg_head, kg_tail, nxt);
    float* tmp = cur; cur = nxt; nxt = tmp;
  }
  // cur = final entity embeddings

  // ---- RotatE rotations + needed masks ----
  k_rfull<<<1, 64, 0, stream>>>(r_vec, rfull);
  k_fmp<<<1, 128, 0, stream>>>(rfull, Fmp);
  hipMemsetAsync(neededU, 0, NG * sizeof(int), stream);
  hipMemsetAsync(neededI, 0, NG * sizeof(int), stream);
  k_needed<<<(BATCH + TB - 1) / TB, TB, 0, stream>>>(tgt_user, neededU);
  k_needed<<<(BATCH + TB - 1) / TB, TB, 0, stream>>>(tgt_item, neededI);

  // ---- 4 metapaths ----
  for (int mp = 0; mp < 4; mp++) {
    const int* idx    = (mp < 2) ? idx_user + (size_t)mp * EMP * 3
                                 : idx_item + (size_t)(mp - 2) * EMP * 3;
    const int* dst    = (mp < 2) ? dst_user + (size_t)mp * EMP
                                 : dst_item + (size_t)(mp - 2) * EMP;
    const int* tgt    = (mp < 2) ? tgt_user : tgt_item;
    const int* need   = (mp < 2) ? neededU : neededI;
    const float* attn = (mp < 2) ? attn_user + (size_t)mp * NHEADS * HID
                                 : attn_item + (size_t)(mp - 2) * NHEADS * HID;
    k_init_sm<<<(NG * NHEADS + TB - 1) / TB, TB, 0, stream>>>(nmax, nsum, NG * NHEADS);
    hipMemsetAsync(ret, 0, (size_t)NG * FEATD * sizeof(float), stream);
    k_mp_edge<<<(EMP + TB - 1) / TB, TB, 0, stream>>>(
        cur, pvs, idx, dst, Fmp + (size_t)mp * 3 * D2 * 2, attn, need, hidden, ebuf, nmax);
    int gEH = (int)(((long long)EMP * NHEADS + TB - 1) / TB);
    k_mp_exp<<<gEH, TB, 0, stream>>>(ebuf, dst, need, nmax, nsum);
    k_mp_scatter<<<gEH, TB, 0, stream>>>(hidden, ebuf, nsum, dst, need, ret);
    k_mp_gather<<<(BATCH * FEATD + TB - 1) / TB, TB, 0, stream>>>(
        ret, tgt, hs + (size_t)mp * BATCH * FEATD);
  }

  // ---- fuse: fc1 GEMMs (WMMA) with fused tanh+colsum, betas, blend ----
  hipMemsetAsync(colsum, 0, 4 * AVD * sizeof(float), stream);
  int wavesFc1 = (BATCH / 16) * (AVD / 16);            // 2048 waves
  int blkFc1 = wavesFc1 * 32 / TB;
  for (int p = 0; p < 4; p++) {
    const float* w1 = (p < 2) ? fc1_w_user : fc1_w_item;
    const float* b1 = (p < 2) ? fc1_b_user : fc1_b_item;
    k_gemm_wmma<<<blkFc1, TB, 0, stream>>>(hs + (size_t)p * BATCH * FEATD, w1, b1,
                                           nullptr, colsum + p * AVD,
                                           BATCH, AVD, FEATD);
  }
  k_betas<<<1, AVD, 0, stream>>>(colsum, fc2_w_user, fc2_w_item, beta);

  float* h_user = out + 2 * (size_t)BATCH * OUTD;
  float* h_item = h_user + (size_t)BATCH * FEATD;
  k_fuse<<<(BATCH * FEATD + TB - 1) / TB, TB, 0, stream>>>(
      hs, hs + (size_t)BATCH * FEATD, beta + 0, h_user);
  k_fuse<<<(BATCH * FEATD + TB - 1) / TB, TB, 0, stream>>>(
      hs + 2 * (size_t)BATCH * FEATD, hs + 3 * (size_t)BATCH * FEATD, beta + 2, h_item);

  // ---- final logits (WMMA) straight into d_out ----
  int wavesLg = (BATCH / 16) * (OUTD / 16);            // 1024 waves
  int blkLg = wavesLg * 32 / TB;
  k_gemm_wmma<<<blkLg, TB, 0, stream>>>(h_user, fc_user_w, fc_user_b,
                                        out, nullptr, BATCH, OUTD, FEATD);
  k_gemm_wmma<<<blkLg, TB, 0, stream>>>(h_item, fc_item_w, fc_item_b,
                                        out + (size_t)BATCH * OUTD, nullptr,
                                        BATCH, OUTD, FEATD);
}